// DCT2_73212012527772
// MI455X (gfx1250) — compile-verified
//
#include <hip/hip_runtime.h>
#include <stddef.h>
#include <stdint.h>

// ---------------------------------------------------------------------------
// 2-D DCT-II (4096x4096 fp32):  out = C * x * C^T,
//   C[u,m] = cos(pi*u*(2m+1)/8192), identical basis for both dims (M==N).
// Reformulated as two *NT* GEMMs so every LDS fragment is k-contiguous:
//   pass 1:  Q = (x * C^T)^T        (NT GEMM, transposed store)
//   pass 2:  out[u][v] = sum_m C[u][m] * Q[v][m]   (NT GEMM, normal store)
// fp32 WMMA (V_WMMA_F32_16X16X4_F32) for full precision; Tensor Data Mover
// DMAs padded 128x32 tiles global->LDS, double-buffered, 1 barrier / K-tile.
// ---------------------------------------------------------------------------

typedef __attribute__((ext_vector_type(2))) float        v2f;
typedef __attribute__((ext_vector_type(4))) float        v4f;
typedef __attribute__((ext_vector_type(8))) float        v8f;
typedef __attribute__((ext_vector_type(4))) unsigned int u32x4;
typedef __attribute__((ext_vector_type(4))) int          i32x4;
typedef __attribute__((ext_vector_type(8))) int          i32x8;

#define MAT  4096
#define BM   128   // block tile rows (A rows)
#define BN   128   // block tile cols (B rows, since NT)
#define BK   32
#define NTHREADS 256
#define LDT  36    // LDS tile row stride: 32 dwords + 4 pad (bank spread)

#if defined(__has_builtin)
#if __has_builtin(__builtin_amdgcn_tensor_load_to_lds)
#define HAVE_TDM 1
#endif
#endif
#ifndef HAVE_TDM
#define HAVE_TDM 0
#endif

// ---------------------------------------------------------------------------
// Exact cosine basis: basis[u*4096+m] = cos(pi*((u*(2m+1)) mod 16384)/8192).
// u*(2m+1) <= 4095*8191 < 2^25 -> exact 32-bit integer phase reduction.
// ---------------------------------------------------------------------------
__global__ __launch_bounds__(NTHREADS)
void dct_basis_kernel(float* __restrict__ basis) {
    unsigned idx = blockIdx.x * NTHREADS + threadIdx.x;
    unsigned u = idx >> 12;
    unsigned m = idx & 4095u;
    unsigned k = (u * (2u * m + 1u)) & 16383u;
    basis[idx] = cospif((float)k * (1.0f / 8192.0f));
}

#if HAVE_TDM
// ---------------------------------------------------------------------------
// TDM 2-D tile load: 128 rows x 32 dwords from a row-major 4096x4096 fp32
// tensor into LDS, padding 4 dwords every 32 dwords -> LDS row stride 36.
// D# per CDNA5 ISA ch.8 (group0: count/lds_addr/global_addr/type=2;
// group1: data_size=4B, pad fields, tensor dims/stride, tile dims).
// ---------------------------------------------------------------------------
__device__ __forceinline__
void tdm_load_tile(const float* gptr, unsigned lds_byte_off) {
    unsigned long long ga = (unsigned long long)(uintptr_t)gptr;

    u32x4 g0;
    g0.x = 1u;                                        // count=1 (valid)
    g0.y = lds_byte_off;                              // lds_addr
    g0.z = (unsigned)ga;                              // global_addr[31:0]
    g0.w = (unsigned)((ga >> 32) & 0x01FFFFFFull)     // global_addr[56:32]
         | (2u << 30);                                // type=2 (image)

    i32x8 g1;
    g1[0] = (int)((2u << 16)                          // data_size = 4B
                | (1u << 20)                          // pad_enable
                | (4u << 22)                          // pad every 32 dwords
                | (3u << 25));                        // pad amount 4 dwords
    g1[1] = (int)(((unsigned)MAT & 0xFFFFu) << 16);   // tensor_dim0 lo16
    g1[2] = (int)(((unsigned)MAT & 0xFFFFu) << 16);   // dim0 hi16(0) | dim1 lo16
    g1[3] = (int)((unsigned)BK << 16);                // dim1 hi16(0) | tile_dim0=32
    g1[4] = (int)BM;                                  // tile_dim1=128 (dim2=0)
    g1[5] = (int)MAT;                                 // tensor_dim0_stride[31:0]
    g1[6] = 0;
    g1[7] = 0;

    i32x4 gz = (i32x4)0;
#if __clang_major__ >= 23
    i32x8 gz8 = (i32x8)0;
    __builtin_amdgcn_tensor_load_to_lds(g0, g1, gz, gz, gz8, 0);
#else
    __builtin_amdgcn_tensor_load_to_lds(g0, g1, gz, gz, 0);
#endif
}
#endif  // HAVE_TDM

// ---------------------------------------------------------------------------
// NT GEMM:  val(row,col) = sum_k A[row][k] * B[col][k]   (both row-major).
// STORE_TRANSPOSED=0: D[row][col] = val  (coalesced b32 stores)
// STORE_TRANSPOSED=1: D[col][row] = val  (per-lane contiguous b128 stores)
// 128x128 block tile, BK=32; 8 wave32s in 4x2 grid; each wave 32x64 =
// 2x4 fp32 16x16 WMMA accumulators -> 64 v_wmma per wave per K-tile.
// ---------------------------------------------------------------------------
template <bool STORE_TRANSPOSED>
__global__ __launch_bounds__(NTHREADS)
void dct_gemm_nt_kernel(const float* __restrict__ A,
                        const float* __restrict__ B,
                        float* __restrict__ D) {
    __shared__ __align__(16) float As[2][BM * LDT];
    __shared__ __align__(16) float Bs[2][BN * LDT];

    const int tid     = threadIdx.x;
    const int lane    = tid & 31;
    const int wave    = tid >> 5;
    const int wr      = wave & 3;          // wave row    (4 x 32 = 128)
    const int wc      = wave >> 2;         // wave column (2 x 64 = 128)
    const int lane16  = lane & 15;
    const int halfSel = lane >> 4;
    const int rowBase = blockIdx.y * BM;
    const int colBase = blockIdx.x * BN;

    v8f acc[2][4] = {};

    // One K-tile: 8 k-steps x 8 WMMAs; all fragments contiguous v2f pairs.
    auto compute_tile = [&](const float* as, const float* bs) {
#pragma unroll
        for (int kk = 0; kk < BK; kk += 4) {
            const int kb = kk + (halfSel << 1);          // fp32 A/B K layout
            v2f a0 = *(const v2f*)&as[(wr * 32 +      lane16) * LDT + kb];
            v2f a1 = *(const v2f*)&as[(wr * 32 + 16 + lane16) * LDT + kb];
            v2f b[4];
#pragma unroll
            for (int j = 0; j < 4; ++j)
                b[j] = *(const v2f*)&bs[(wc * 64 + j * 16 + lane16) * LDT + kb];
#pragma unroll
            for (int j = 0; j < 4; ++j) {
                acc[0][j] = __builtin_amdgcn_wmma_f32_16x16x4_f32(
                    false, a0, false, b[j], (short)0, acc[0][j], false, false);
                acc[1][j] = __builtin_amdgcn_wmma_f32_16x16x4_f32(
                    false, a1, false, b[j], (short)0, acc[1][j], false, false);
            }
        }
    };

#if HAVE_TDM
    const unsigned asOff[2] = { (unsigned)(uintptr_t)&As[0][0],
                                (unsigned)(uintptr_t)&As[1][0] };
    const unsigned bsOff[2] = { (unsigned)(uintptr_t)&Bs[0][0],
                                (unsigned)(uintptr_t)&Bs[1][0] };

    // Prologue: wave 0 DMAs K-tile 0 into buffer 0.
    if (wave == 0) {
        tdm_load_tile(A + (size_t)rowBase * MAT, asOff[0]);
        tdm_load_tile(B + (size_t)colBase * MAT, bsOff[0]);
    }

    for (int kt = 0; kt < MAT; kt += BK) {
        const int buf = (kt >> 5) & 1;
        if (wave == 0) __builtin_amdgcn_s_wait_tensorcnt((short)0);
        __syncthreads();   // tile `buf` ready; all waves done with `buf^1`
        if (wave == 0 && (kt + BK) < MAT) {
            const int nb = buf ^ 1;
            tdm_load_tile(A + (size_t)rowBase * MAT + (kt + BK), asOff[nb]);
            tdm_load_tile(B + (size_t)colBase * MAT + (kt + BK), bsOff[nb]);
        }
        compute_tile(&As[buf][0], &Bs[buf][0]);   // overlaps next tile's DMA
    }
#else
    // Fallback: manual global->LDS staging, single buffer.
    for (int kt = 0; kt < MAT; kt += BK) {
#pragma unroll
        for (int i = 0; i < 4; ++i) {
            int linear = tid + i * NTHREADS;       // 0..1023
            int r = linear >> 3;                   // 0..127
            int c = (linear & 7) << 2;             // 0..28
            v4f va = *(const v4f*)(A + (size_t)(rowBase + r) * MAT + kt + c);
            *(v4f*)(&As[0][r * LDT + c]) = va;
            v4f vb = *(const v4f*)(B + (size_t)(colBase + r) * MAT + kt + c);
            *(v4f*)(&Bs[0][r * LDT + c]) = vb;
        }
        __syncthreads();
        compute_tile(&As[0][0], &Bs[0][0]);
        __syncthreads();
    }
#endif

    // Epilogue. C/D layout: VGPR r holds rows r (lanes 0-15) / r+8 (16-31),
    // so per lane the 8 accumulator elements are 8 *consecutive* rows.
#pragma unroll
    for (int i = 0; i < 2; ++i) {
#pragma unroll
        for (int j = 0; j < 4; ++j) {
            const int row0 = rowBase + wr * 32 + i * 16 + (halfSel << 3);
            const int col  = colBase + wc * 64 + j * 16 + lane16;
            if (STORE_TRANSPOSED) {
                // D[col][row0 .. row0+7]: two aligned b128 stores per tile.
                v4f lo = __builtin_shufflevector(acc[i][j], acc[i][j], 0, 1, 2, 3);
                v4f hi = __builtin_shufflevector(acc[i][j], acc[i][j], 4, 5, 6, 7);
                *(v4f*)(D + (size_t)col * MAT + row0    ) = lo;
                *(v4f*)(D + (size_t)col * MAT + row0 + 4) = hi;
            } else {
                // D[row][col]: lane16 gives 64B-coalesced b32 stores.
#pragma unroll
                for (int r = 0; r < 8; ++r)
                    D[(size_t)(row0 + r) * MAT + col] = acc[i][j][r];
            }
        }
    }
}

// ---------------------------------------------------------------------------
// Launch:
//   basis build; Q = (x * C^T)^T; out = C * Q^T  -> out = C * x * C^T.
// Workspace: [0,64MB) = basis C, [64MB,128MB) = Q.
// ---------------------------------------------------------------------------
extern "C" void kernel_launch(void* const* d_in, const int* in_sizes, int n_in,
                              void* d_out, int out_size, void* d_ws, size_t ws_size,
                              hipStream_t stream) {
    (void)in_sizes; (void)n_in; (void)out_size; (void)ws_size;

    const float* x     = (const float*)d_in[0];
    float*       out   = (float*)d_out;
    float*       basis = (float*)d_ws;
    float*       Q     = basis + (size_t)MAT * MAT;

    dct_basis_kernel<<<(MAT * MAT) / NTHREADS, NTHREADS, 0, stream>>>(basis);

    dim3 grid(MAT / BN, MAT / BM), block(NTHREADS);
    // pass 1: Q[v][m] = sum_n x[m][n] * C[v][n]   (NT, transposed store)
    dct_gemm_nt_kernel<true ><<<grid, block, 0, stream>>>(x, basis, Q);
    // pass 2: out[u][v] = sum_m C[u][m] * Q[v][m] (NT, normal store)
    dct_gemm_nt_kernel<false><<<grid, block, 0, stream>>>(basis, Q, out);
}